// Attack_net_withlabelsL2_67654324847187
// MI455X (gfx1250) — compile-verified
//
#include <hip/hip_runtime.h>

// ---------------------------------------------------------------------------
// MoE attack-net forward for MI455X (gfx1250, wave32).
//  - fp32 WMMA (v_wmma_f32_16x16x4_f32) for the dominant encoder GEMM,
//    two 16-row tiles fused per wave + software-pipelined LDS fragment loads
//  - all weights resident in LDS (311 KB of the 320 KB/WGP)
//  - per-sample expert routing (avoids the reference's 10x dense decoder)
// ---------------------------------------------------------------------------

typedef float v2f __attribute__((ext_vector_type(2)));
typedef float v8f __attribute__((ext_vector_type(8)));

#define N_TOTAL        262144
#define NCLS           10
#define SLOPE          0.01f
#define EPS_R          0.5f

#define BLOCK_THREADS  128           // 4 wave32
#define CHUNK          128           // samples per chunk = 8 row-tiles of 16
#define CHUNKS_PER_BLK 8
#define SAMPLES_PER_BLK (CHUNK * CHUNKS_PER_BLK)   // 1024
#define GRID_BLOCKS    (N_TOTAL / SAMPLES_PER_BLK) // 256

#define K1P      52    // encoder hidden 50 padded to multiple of 4
#define N2P      112   // encoder output 100 padded to multiple of 16
#define HSTRIDE  104   // LDS row stride for staged activations (8B aligned rows)
#define KSTEPS   13    // 52 / 4

// LDS layout (float offsets)
enum {
  OFF_W1  = 0,                     // enc_W1   [2][50]
  OFF_B1  = OFF_W1  + 100,         // enc_b1   [50]
  OFF_W2T = OFF_B1  + 50,          // enc_W2^T [112][52]  (zero padded)
  OFF_B2  = OFF_W2T + N2P * K1P,   // enc_b2   [112]      (zero padded)
  OFF_DW1 = OFF_B2  + N2P,         // dec_W1   [10][100][50]
  OFF_DB1 = OFF_DW1 + NCLS*100*50, // dec_b1   [10][50]
  OFF_DW2 = OFF_DB1 + NCLS*50,     // dec_W2   [10][50][15]
  OFF_DB2 = OFF_DW2 + NCLS*50*15,  // dec_b2   [10][15]
  OFF_DW3 = OFF_DB2 + NCLS*15,     // dec_W3   [10][15][2]
  OFF_DB3 = OFF_DW3 + NCLS*15*2,   // dec_b3   [10][2]
  OFF_H   = OFF_DB3 + NCLS*2,      // staged activations [128][104]
  LDS_FLOATS = OFF_H + CHUNK * HSTRIDE
};
#define LDS_BYTES (LDS_FLOATS * 4)   // 311,472 B < 320 KB/WGP

__device__ __forceinline__ float lrelu(float v) {
  return v > 0.0f ? v : SLOPE * v;
}

__device__ __forceinline__ void copy_lds(float* dst, const float* src, int n,
                                         int tid) {
  for (int i = tid; i < n; i += BLOCK_THREADS) dst[i] = src[i];
}

__global__ __launch_bounds__(BLOCK_THREADS)
void attack_net_moe_kernel(const float* __restrict__ x,
                           const int*   __restrict__ y,
                           const float* __restrict__ enc_W1,
                           const float* __restrict__ enc_b1,
                           const float* __restrict__ enc_W2,
                           const float* __restrict__ enc_b2,
                           const float* __restrict__ dec_W1,
                           const float* __restrict__ dec_b1,
                           const float* __restrict__ dec_W2,
                           const float* __restrict__ dec_b2,
                           const float* __restrict__ dec_W3,
                           const float* __restrict__ dec_b3,
                           float* __restrict__ out) {
  extern __shared__ float sm[];
  const int tid  = threadIdx.x;
  const int lane = tid & 31;
  const int wv   = tid >> 5;       // wave id 0..3
  const int lm   = lane & 15;      // lane 0..15 within half-wave
  const int lg   = lane >> 4;      // half-wave select (K msb / row+8)

  // ---- one-time: stage all weights into LDS -------------------------------
  copy_lds(&sm[OFF_W1],  enc_W1, 100,          tid);
  copy_lds(&sm[OFF_B1],  enc_b1, 50,           tid);
  copy_lds(&sm[OFF_DW1], dec_W1, NCLS*100*50,  tid);
  copy_lds(&sm[OFF_DB1], dec_b1, NCLS*50,      tid);
  copy_lds(&sm[OFF_DW2], dec_W2, NCLS*50*15,   tid);
  copy_lds(&sm[OFF_DB2], dec_b2, NCLS*15,      tid);
  copy_lds(&sm[OFF_DW3], dec_W3, NCLS*15*2,    tid);
  copy_lds(&sm[OFF_DB3], dec_b3, NCLS*2,       tid);
  // enc_W2 (50x100) -> transposed, zero-padded [112][52] so B fragments are
  // contiguous 8-byte LDS loads (ds_load_b64 with immediate offsets).
  for (int i = tid; i < N2P * K1P; i += BLOCK_THREADS) {
    const int n = i / K1P, k = i - n * K1P;
    sm[OFF_W2T + i] = (n < 100 && k < 50) ? enc_W2[k * 100 + n] : 0.0f;
  }
  for (int i = tid; i < N2P; i += BLOCK_THREADS)
    sm[OFF_B2 + i] = (i < 100) ? enc_b2[i] : 0.0f;
  __syncthreads();

  const int blockBase = blockIdx.x * SAMPLES_PER_BLK;

  // Base addresses for fragment loads (8-byte aligned -> v2f == ds_load_b64).
  // A frag (f32 16x4): lane holds row lm, K = 4k + 2*lg + {0,1}
  // B frag (4x16)    : lane holds col lm, K = 4k + 2*lg + {0,1}  (W2T layout)
  const float* aBase0 = &sm[OFF_H + (wv * 32 + lm) * HSTRIDE + lg * 2];
  const float* aBase1 = aBase0 + 16 * HSTRIDE;
  const float* bBase  = &sm[OFF_W2T + lm * K1P + lg * 2];

  for (int chunk = 0; chunk < CHUNKS_PER_BLK; ++chunk) {
    const int base = blockBase + chunk * CHUNK;
    const int g    = base + tid;              // this thread's sample

    // ---- layer 1 (K=2): per-thread VALU, stage h1 to LDS ------------------
    {
      const float x0 = x[2 * g];
      const float x1 = x[2 * g + 1];
      float* hrow = &sm[OFF_H + tid * HSTRIDE];
      #pragma unroll 10
      for (int j = 0; j < 50; ++j) {
        float v = fmaf(x0, sm[OFF_W1 + j],
                  fmaf(x1, sm[OFF_W1 + 50 + j], sm[OFF_B1 + j]));
        hrow[j] = lrelu(v);
      }
      hrow[50] = 0.0f;   // K padding
      hrow[51] = 0.0f;
    }
    // prefetch next chunk's inputs while we crunch
    if (chunk + 1 < CHUNKS_PER_BLK) {
      __builtin_prefetch(&x[2 * (g + CHUNK)], 0, 1);
    }
    __syncthreads();

    // ---- layer 2: [32x52] x [52x112] via v_wmma_f32_16x16x4_f32 -----------
    // Wave w owns rows [32w, 32w+32): both 16-row tiles share B fragments,
    // so fuse them in one K loop (14 WMMAs per K-step) and software-pipeline
    // the 9 LDS fragment loads one K-step ahead.
    {
      v8f acc0[7], acc1[7];
      const v8f zero = {0.f, 0.f, 0.f, 0.f, 0.f, 0.f, 0.f, 0.f};
      #pragma unroll
      for (int nt = 0; nt < 7; ++nt) { acc0[nt] = zero; acc1[nt] = zero; }

      v2f a0 = *(const v2f*)(aBase0);
      v2f a1 = *(const v2f*)(aBase1);
      v2f b[7];
      #pragma unroll
      for (int nt = 0; nt < 7; ++nt)
        b[nt] = *(const v2f*)(bBase + nt * 16 * K1P);

      #pragma unroll
      for (int k = 0; k < KSTEPS; ++k) {
        v2f a0n = a0, a1n = a1, bn[7];
        #pragma unroll
        for (int nt = 0; nt < 7; ++nt) bn[nt] = b[nt];
        if (k + 1 < KSTEPS) {
          const int kb = (k + 1) * 4;
          a0n = *(const v2f*)(aBase0 + kb);
          a1n = *(const v2f*)(aBase1 + kb);
          #pragma unroll
          for (int nt = 0; nt < 7; ++nt)
            bn[nt] = *(const v2f*)(bBase + nt * 16 * K1P + kb);
        }
        #pragma unroll
        for (int nt = 0; nt < 7; ++nt) {
          acc0[nt] = __builtin_amdgcn_wmma_f32_16x16x4_f32(
              false, a0, false, b[nt], (short)0, acc0[nt], false, false);
          acc1[nt] = __builtin_amdgcn_wmma_f32_16x16x4_f32(
              false, a1, false, b[nt], (short)0, acc1[nt], false, false);
        }
        a0 = a0n; a1 = a1n;
        #pragma unroll
        for (int nt = 0; nt < 7; ++nt) b[nt] = bn[nt];
      }

      // write back with bias + LeakyReLU (C layout: vgpr r -> M=r(+8), N=lane)
      #pragma unroll
      for (int nt = 0; nt < 7; ++nt) {
        const int col = nt * 16 + lm;
        if (col < 100) {
          const float bias = sm[OFF_B2 + col];
          #pragma unroll
          for (int r = 0; r < 8; ++r) {
            const int row = wv * 32 + r + lg * 8;
            sm[OFF_H + row * HSTRIDE + col]        = lrelu(acc0[nt][r] + bias);
            sm[OFF_H + (row + 16) * HSTRIDE + col] = lrelu(acc1[nt][r] + bias);
          }
        }
      }
    }
    __syncthreads();

    // ---- routed decoder: one thread per sample ----------------------------
    {
      const int cls = y[g];
      const float* hrow = &sm[OFF_H + tid * HSTRIDE];

      float a1v[50];
      #pragma unroll
      for (int o = 0; o < 50; ++o) a1v[o] = sm[OFF_DB1 + cls * 50 + o];
      for (int d = 0; d < 100; ++d) {
        const float hd = hrow[d];
        const float* w = &sm[OFF_DW1 + (cls * 100 + d) * 50];
        #pragma unroll
        for (int o = 0; o < 50; ++o) a1v[o] = fmaf(hd, w[o], a1v[o]);
      }
      #pragma unroll
      for (int o = 0; o < 50; ++o) a1v[o] = lrelu(a1v[o]);

      float a2v[15];
      #pragma unroll
      for (int o = 0; o < 15; ++o) a2v[o] = sm[OFF_DB2 + cls * 15 + o];
      for (int d = 0; d < 50; ++d) {
        const float v = a1v[d];
        const float* w = &sm[OFF_DW2 + (cls * 50 + d) * 15];
        #pragma unroll
        for (int o = 0; o < 15; ++o) a2v[o] = fmaf(v, w[o], a2v[o]);
      }
      #pragma unroll
      for (int o = 0; o < 15; ++o) a2v[o] = lrelu(a2v[o]);

      float s0 = sm[OFF_DB3 + cls * 2 + 0];
      float s1 = sm[OFF_DB3 + cls * 2 + 1];
      #pragma unroll
      for (int d = 0; d < 15; ++d) {
        const float v = a2v[d];
        s0 = fmaf(v, sm[OFF_DW3 + (cls * 15 + d) * 2 + 0], s0);
        s1 = fmaf(v, sm[OFF_DW3 + (cls * 15 + d) * 2 + 1], s1);
      }

      const float inv = rsqrtf(fmaf(s0, s0, s1 * s1));
      out[2 * g]     = EPS_R * s0 * inv;
      out[2 * g + 1] = EPS_R * s1 * inv;
    }
    __syncthreads();   // hStage reused by next chunk
  }
}

extern "C" void kernel_launch(void* const* d_in, const int* in_sizes, int n_in,
                              void* d_out, int out_size, void* d_ws, size_t ws_size,
                              hipStream_t stream) {
  (void)in_sizes; (void)n_in; (void)out_size; (void)d_ws; (void)ws_size;
  const float* x      = (const float*)d_in[0];
  const int*   y      = (const int*)  d_in[1];
  const float* enc_W1 = (const float*)d_in[2];
  const float* enc_b1 = (const float*)d_in[3];
  const float* enc_W2 = (const float*)d_in[4];
  const float* enc_b2 = (const float*)d_in[5];
  const float* dec_W1 = (const float*)d_in[6];
  const float* dec_b1 = (const float*)d_in[7];
  const float* dec_W2 = (const float*)d_in[8];
  const float* dec_b2 = (const float*)d_in[9];
  const float* dec_W3 = (const float*)d_in[10];
  const float* dec_b3 = (const float*)d_in[11];
  float* out = (float*)d_out;

  (void)hipFuncSetAttribute((const void*)attack_net_moe_kernel,
                            hipFuncAttributeMaxDynamicSharedMemorySize,
                            LDS_BYTES);
  attack_net_moe_kernel<<<GRID_BLOCKS, BLOCK_THREADS, LDS_BYTES, stream>>>(
      x, y, enc_W1, enc_b1, enc_W2, enc_b2,
      dec_W1, dec_b1, dec_W2, dec_b2, dec_W3, dec_b3, out);
}